// NeRFLoss_82480551952463
// MI455X (gfx1250) — compile-verified
//
#include <hip/hip_runtime.h>
#include <math.h>

#define N_RAYS 65536
#define SAMP   128
#define NTOT   (N_RAYS * SAMP)

typedef __attribute__((ext_vector_type(2))) float v2f;
typedef __attribute__((ext_vector_type(8))) float v8f;
typedef __attribute__((ext_vector_type(4))) int   v4i;

#if defined(__HIP_DEVICE_COMPILE__) &&                                   \
    __has_builtin(__builtin_amdgcn_global_load_async_to_lds_b128) &&     \
    __has_builtin(__builtin_amdgcn_s_wait_asynccnt)
#define USE_ASYNC_LDS 1
#else
#define USE_ASYNC_LDS 0
#endif

// ---------------------------------------------------------------------------
// Kernel 1: grid-stride partial sums for the 5 scalar losses.
// partials layout: partials[block*5 + q]
// ---------------------------------------------------------------------------
__global__ __launch_bounds__(256) void scalar_partials(
    const float* __restrict__ rgb,  const float* __restrict__ trgb,
    const float* __restrict__ sw,   const float* __restrict__ op,
    const float* __restrict__ opd,  const float* __restrict__ se,
    float* __restrict__ partials)
{
    float a0 = 0.f, a1 = 0.f, a2 = 0.f, a3 = 0.f, a4 = 0.f;
    const int tid    = blockIdx.x * blockDim.x + threadIdx.x;
    const int stride = gridDim.x * blockDim.x;
    for (int i = tid; i < NTOT; i += stride) {
        float y = sw[i];
        y = fminf(fmaxf(y, 1e-7f), 1.0f);
        a4 += -y * logf(y);
        a3 += se[i];
        if (i < 3 * N_RAYS) { float d = rgb[i] - trgb[i]; a0 += d * d; }
        if (i < N_RAYS) {
            float o  = op[i]  + 1e-10f; a1 += -o  * logf(o);
            float od = opd[i] + 1e-10f; a2 += -od * logf(od);
        }
    }
    __shared__ float red[256];
    float acc[5] = {a0, a1, a2, a3, a4};
    for (int q = 0; q < 5; q++) {
        red[threadIdx.x] = acc[q];
        __syncthreads();
        for (int off = 128; off > 0; off >>= 1) {
            if ((int)threadIdx.x < off) red[threadIdx.x] += red[threadIdx.x + off];
            __syncthreads();
        }
        if (threadIdx.x == 0) partials[blockIdx.x * 5 + q] = red[0];
        __syncthreads();
    }
}

// ---------------------------------------------------------------------------
// Kernel 2: deterministic final reduction + scaling. out[0..4].
// Output order: [d_rgb, d_opacity, d_opacity_dyn, d_sigma_entropy, d_entropy]
// ---------------------------------------------------------------------------
__global__ void scalar_final(const float* __restrict__ partials, int nblocks,
                             float* __restrict__ out)
{
    int q = threadIdx.x;
    if (q < 5) {
        double s = 0.0;
        for (int b = 0; b < nblocks; b++) s += (double)partials[b * 5 + q];
        float r;
        switch (q) {
            case 0:  r = (float)(s / (3.0 * N_RAYS));         break; // d_rgb
            case 1:  r = (float)(s / N_RAYS * 1e-3);          break; // d_opacity
            case 2:  r = (float)(s / N_RAYS * 1e-10);         break; // d_opacity_dyn
            case 3:  r = (float)(s / (double)NTOT * 1e-7);    break; // d_sigma_entropy
            default: r = (float)(s / (double)NTOT * 1e-3);    break; // d_entropy
        }
        out[q] = r;
    }
}

// ---------------------------------------------------------------------------
// Kernel 3: distortion loss. One wave32 handles 2 rays (256 samples) as a
// 16x16 chunk tile.  Within-chunk exclusive scans of w and w*t are computed
// with V_WMMA_F32_16X16X4_F32 against a strictly-upper-triangular 0/1 matrix
// (D[c][j] = sum_{k<j} W[c][k]), chained 4x over K.  D-layout puts ray0 in
// lanes 0-15 and ray1 in lanes 16-31; cross-chunk offsets are 8 serial adds.
// Staging uses GLOBAL_LOAD_ASYNC_TO_LDS_B128 (memory->LDS direct, ASYNCcnt)
// when the toolchain exposes the builtin; wave-private data means a per-wave
// s_wait_asynccnt replaces any block barrier.
// ---------------------------------------------------------------------------
__global__ __launch_bounds__(256) void distortion_wmma(
    const float* __restrict__ wsg,
    const float* __restrict__ dlg,
    const float* __restrict__ tsg,
    float* __restrict__ out)
{
    __shared__ float sW[8 * 256];
    __shared__ float sT[8 * 256];
    __shared__ float sD[8 * 256];

    const int wv   = threadIdx.x >> 5;
    const int lane = threadIdx.x & 31;
    const int half = lane >> 4;     // 0: ray0 rows / K lo, 1: ray1 rows / K hi
    const int j    = lane & 15;

    const int  pairIdx = blockIdx.x * 8 + wv;      // 0..32767, 2 rays each
    const long base    = (long)pairIdx * 256;

    float* w  = &sW[wv * 256];
    float* t  = &sT[wv * 256];
    float* dl = &sD[wv * 256];

#if USE_ASYNC_LDS
    {
        // Each lane covers 16B; one issue covers 512B, imm offset covers the
        // second 512B (offset is added to BOTH global and LDS addresses).
        typedef __attribute__((address_space(1))) v4i gv4i;  // global int4
        typedef __attribute__((address_space(3))) v4i lv4i;  // LDS int4
        gv4i* gW = (gv4i*)(wsg + base + 4 * lane);
        gv4i* gT = (gv4i*)(tsg + base + 4 * lane);
        gv4i* gD = (gv4i*)(dlg + base + 4 * lane);
        lv4i* lW = (lv4i*)(w  + 4 * lane);
        lv4i* lT = (lv4i*)(t  + 4 * lane);
        lv4i* lD = (lv4i*)(dl + 4 * lane);
        __builtin_amdgcn_global_load_async_to_lds_b128(gW, lW, 0,   0);
        __builtin_amdgcn_global_load_async_to_lds_b128(gW, lW, 512, 0);
        __builtin_amdgcn_global_load_async_to_lds_b128(gT, lT, 0,   0);
        __builtin_amdgcn_global_load_async_to_lds_b128(gT, lT, 512, 0);
        __builtin_amdgcn_global_load_async_to_lds_b128(gD, lD, 0,   0);
        __builtin_amdgcn_global_load_async_to_lds_b128(gD, lD, 512, 0);
        __builtin_amdgcn_s_wait_asynccnt(0);
        asm volatile("" ::: "memory");   // keep LDS reads below the wait
    }
#else
    #pragma unroll
    for (int k = 0; k < 8; k++) {                  // coalesced staging
        int li = k * 32 + lane;
        w[li]  = wsg[base + li];
        t[li]  = tsg[base + li];
        dl[li] = dlg[base + li];
    }
    __syncthreads();
#endif

    // Within-chunk exclusive scans via WMMA: acc = W(16x16) x U(16x16,k<j)
    v8f accW  = {};
    v8f accWT = {};
    #pragma unroll
    for (int s = 0; s < 4; s++) {
        const int kb = 4 * s + 2 * half;           // A/B K rows for this half
        v2f aW, aWT, bm;
        float w0 = w[j * 16 + kb + 0];
        float w1 = w[j * 16 + kb + 1];
        aW.x  = w0;                    aW.y  = w1;
        aWT.x = w0 * t[j * 16 + kb];   aWT.y = w1 * t[j * 16 + kb + 1];
        bm.x  = (kb + 0 < j) ? 1.0f : 0.0f;        // strictly upper triangular
        bm.y  = (kb + 1 < j) ? 1.0f : 0.0f;
        accW  = __builtin_amdgcn_wmma_f32_16x16x4_f32(false, aW,  false, bm,
                                                      (short)0, accW,  false, false);
        accWT = __builtin_amdgcn_wmma_f32_16x16x4_f32(false, aWT, false, bm,
                                                      (short)0, accWT, false, false);
    }

    // Per-ray exclusive chunk offsets (8 chunks). Lane 15/31 column holds the
    // last within-chunk exclusive value; add last element for inclusive total.
    float offW[8], offWT[8];
    float runW = 0.f, runWT = 0.f;
    #pragma unroll
    for (int p = 0; p < 8; p++) {
        offW[p]  = runW;
        offWT[p] = runWT;
        int   last = (p + 8 * half) * 16 + 15;
        float wl   = w[last];
        float totW  = __shfl(accW[p],  15, 16) + wl;
        float totWT = __shfl(accWT[p], 15, 16) + wl * t[last];
        runW  += totW;
        runWT += totWT;
    }

    // per-sample: 2 w (t*A - B) + w^2 * delta / 3 ; then per-ray sum
    float sum = 0.f;
    #pragma unroll
    for (int p = 0; p < 8; p++) {
        int   sidx = (p + 8 * half) * 16 + j;
        float wi = w[sidx], ti = t[sidx], di = dl[sidx];
        float A = accW[p]  + offW[p];
        float B = accWT[p] + offWT[p];
        sum += 2.0f * wi * (ti * A - B) + wi * wi * di * (1.0f / 3.0f);
    }
    #pragma unroll
    for (int off = 8; off > 0; off >>= 1) sum += __shfl_xor(sum, off, 16);

    if (j == 0) {
        int ray = pairIdx * 2 + half;
        out[5 + ray] = 0.001f * sum;               // LAMBDA_DISTORTION
    }
}

// ---------------------------------------------------------------------------
extern "C" void kernel_launch(void* const* d_in, const int* in_sizes, int n_in,
                              void* d_out, int out_size, void* d_ws, size_t ws_size,
                              hipStream_t stream)
{
    const float* rgb  = (const float*)d_in[0];
    const float* trgb = (const float*)d_in[1];
    const float* sw   = (const float*)d_in[2];
    const float* op   = (const float*)d_in[3];
    const float* opd  = (const float*)d_in[4];
    const float* se   = (const float*)d_in[5];
    const float* wsp  = (const float*)d_in[6];
    const float* dls  = (const float*)d_in[7];
    const float* tsp  = (const float*)d_in[8];
    // d_in[9] = rays_a: layout is fixed (ray r starts at r*128), not needed.

    float* out      = (float*)d_out;
    float* partials = (float*)d_ws;                // 1024*5 floats = 20 KB

    const int B1 = 1024;
    scalar_partials<<<B1, 256, 0, stream>>>(rgb, trgb, sw, op, opd, se, partials);
    scalar_final<<<1, 32, 0, stream>>>(partials, B1, out);
    distortion_wmma<<<N_RAYS / 16, 256, 0, stream>>>(wsp, dls, tsp, out);
}